// BivectorPhasorBlock_84825604096579
// MI455X (gfx1250) — compile-verified
//
#include <hip/hip_runtime.h>
#include <math.h>

// ---------------------------------------------------------------------------
// BivectorPhasorBlock on MI455X (gfx1250, wave32)
// B=2, L=4096, D=1024, K=256.  All math fp32, matching the reference.
// GEMMs: V_WMMA_F32_16X16X4_F32 + double-buffered GLOBAL_LOAD_ASYNC_TO_LDS_B128
// (ASYNCcnt) so next-tile global traffic overlaps current-tile WMMA compute.
// ---------------------------------------------------------------------------

typedef __attribute__((ext_vector_type(2))) float v2f;
typedef __attribute__((ext_vector_type(8))) float v8f;

#define HALF_PI 1.5707963267948966f

enum { ACT_NONE = 0, ACT_GELU = 1, ACT_TANHPI = 2 };

__device__ __forceinline__ float act_gelu_exact(float x) {
    // torch nn.GELU default (exact): 0.5*x*(1+erf(x/sqrt(2)))
    return 0.5f * x * (1.0f + erff(x * 0.7071067811865476f));
}

// Async global->LDS copy of 16 bytes. VDST operand carries the LDS byte
// offset, VADDR the 64-bit global address (GV mode). Tracked by ASYNCcnt.
__device__ __forceinline__ void async_copy_b128(unsigned lds_off, const float* gaddr) {
    asm volatile("global_load_async_to_lds_b128 %0, %1, off"
                 :
                 : "v"(lds_off), "v"(gaddr)
                 : "memory");
}
__device__ __forceinline__ void wait_async_zero() {
    asm volatile("s_wait_asynccnt 0x0" ::: "memory");
}

// LDS byte offset of a generic (__shared__) pointer: low 32 bits of the
// generic address are the LDS offset (aperture mapping, ISA 10.2).
__device__ __forceinline__ unsigned lds_offset(const void* p) {
    return (unsigned)(uintptr_t)p;
}

// ---------------------------------------------------------------------------
// Tiled fp32 WMMA GEMM:  C[M,N] = epilogue(A[M,Kd] @ W[Kd,N] + bias[N])
// Block tile 64x64, 128 threads (4 waves). Each wave owns a 16-row strip and
// 4 WMMA 16x16 C tiles across N. K staged through double-buffered LDS in
// steps of 16 (4 x wmma_f32_16x16x4_f32 per n-tile per step).
// A-tile LDS stride 20 floats: rows 16B-aligned (b128 staging), fragments
// 8B-aligned (single ds_load_b64), 20*i mod 64 distinct -> conflict-free.
// B-tile LDS stride 72 floats: rows 16B-aligned, fragment reads consecutive
// columns -> conflict-free.
// Requires M%64==0, N%64==0, Kd%16==0 (true for all six GEMMs here).
// ---------------------------------------------------------------------------
template <int ACT, bool RESID>
__global__ __launch_bounds__(128)
void gemm_wmma_f32(const float* __restrict__ A, const float* __restrict__ W,
                   const float* __restrict__ bias, const float* __restrict__ resid,
                   float* __restrict__ C, int M, int N, int Kd) {
    __shared__ float As[2][64][20];
    __shared__ float Bs[2][16][72];

    const int tid   = threadIdx.x;
    const int wave  = tid >> 5;
    const int lane  = tid & 31;
    const int laneM = lane & 15;  // row/col within 16
    const int laneH = lane >> 4;  // half-wave select
    const int mBase = blockIdx.y * 64;
    const int nBase = blockIdx.x * 64;
    const int wm    = wave * 16;  // wave's row strip inside block tile

    v8f acc[4] = {};

    // Issue the async copies for one K-tile into LDS buffer `buf`.
    // 256 b128 chunks for A (64 rows x 64B) + 256 for B (16 rows x 256B),
    // 2 + 2 chunks per thread.
    auto issue_tile = [&](int buf, int k0) {
        const unsigned aBase = lds_offset(&As[buf][0][0]);
        const unsigned bBase = lds_offset(&Bs[buf][0][0]);
#pragma unroll
        for (int i = 0; i < 2; ++i) {
            const int slot = tid * 2 + i;  // 0..255
            {   // A tile 64x16
                const int r  = slot >> 2;
                const int c4 = slot & 3;
                const unsigned loff = aBase + (unsigned)((r * 20 + c4 * 4) * 4);
                async_copy_b128(loff, A + (size_t)(mBase + r) * Kd + k0 + c4 * 4);
            }
            {   // B tile 16x64
                const int r  = slot >> 4;
                const int c4 = slot & 15;
                const unsigned loff = bBase + (unsigned)((r * 72 + c4 * 4) * 4);
                async_copy_b128(loff, W + (size_t)(k0 + r) * N + nBase + c4 * 4);
            }
        }
    };

    auto compute_tile = [&](int buf) {
#pragma unroll
        for (int kk = 0; kk < 4; ++kk) {
            // A fragment 16x4 (ISA 7.12.2): lanes 0-15 K={0,1}, lanes 16-31
            // K={2,3}; contiguous + 8B aligned -> single ds_load_b64.
            const int kr = kk * 4 + 2 * laneH;
            const v2f a = *(const v2f*)&As[buf][wm + laneM][kr];
#pragma unroll
            for (int j = 0; j < 4; ++j) {
                // B fragment 4x16: rows striped across lanes, mirror of A.
                v2f b;
                b.x = Bs[buf][kr + 0][j * 16 + laneM];
                b.y = Bs[buf][kr + 1][j * 16 + laneM];
                acc[j] = __builtin_amdgcn_wmma_f32_16x16x4_f32(
                    /*neg_a=*/false, a, /*neg_b=*/false, b,
                    /*c_mod=*/(short)0, acc[j],
                    /*reuse_a=*/false, /*reuse_b=*/false);
            }
        }
    };

    // Software pipeline: prologue tile 0, then overlap async(k+16) with wmma(k).
    int buf = 0;
    issue_tile(0, 0);
    wait_async_zero();
    __syncthreads();
    for (int k0 = 0; k0 < Kd; k0 += 16) {
        if (k0 + 16 < Kd) issue_tile(buf ^ 1, k0 + 16);
        compute_tile(buf);
        wait_async_zero();   // own wave's async writes landed in LDS
        __syncthreads();     // all waves done computing + all copies visible
        buf ^= 1;
    }

    // ---- epilogue: bias + activation (+ residual), C layout per ISA 7.12.2 ----
#pragma unroll
    for (int j = 0; j < 4; ++j) {
#pragma unroll
        for (int r = 0; r < 8; ++r) {
            const int row = mBase + wm + laneH * 8 + r;
            const int col = nBase + j * 16 + laneM;
            float v = acc[j][r] + bias[col];
            if (ACT == ACT_GELU)        v = act_gelu_exact(v);
            else if (ACT == ACT_TANHPI) v = tanhf(v) * HALF_PI;
            if (RESID) v += resid[(size_t)row * N + col];
            C[(size_t)row * N + col] = v;
        }
    }
}

// ---------------------------------------------------------------------------
// Rotor helpers (matches reference _rotor/_apply_rotor)
// ---------------------------------------------------------------------------
__device__ __forceinline__ void apply_rotor(float s, const float* t,
                                            float v1, float v2, float v3, float v4,
                                            float& o1, float& o2, float& o3, float& o4) {
    const float t12 = t[0], t13 = t[1], t14 = t[2];
    const float t23 = t[3], t24 = t[4], t34 = t[5];
    const float s2 = s * s, ts = 2.0f * s;
    o1 = v1 * (s2 - t12 * t12 - t13 * t13 - t14 * t14) + ts * ( v2 * t12 + v3 * t13 + v4 * t14);
    o2 = v2 * (s2 - t12 * t12 - t23 * t23 - t24 * t24) + ts * (-v1 * t12 + v3 * t23 + v4 * t24);
    o3 = v3 * (s2 - t13 * t13 - t23 * t23 - t34 * t34) + ts * (-v1 * t13 - v2 * t23 + v4 * t34);
    o4 = v4 * (s2 - t14 * t14 - t24 * t24 - t34 * t34) + ts * (-v1 * t14 - v2 * t24 - v3 * t34);
}

// Key rotor applied to value, in place on v (one thread per (b,l,k) group).
__global__ __launch_bounds__(256)
void rotor_key_apply(const float* __restrict__ ang, float* __restrict__ v, int total) {
    const int idx = blockIdx.x * 256 + threadIdx.x;
    if (idx >= total) return;
    const float* a = ang + (size_t)idx * 6;
    float t[6];
    float m2 = 0.f;
#pragma unroll
    for (int i = 0; i < 6; ++i) { t[i] = a[i]; m2 += t[i] * t[i]; }
    const float mag  = fmaxf(sqrtf(m2), 1e-8f);
    const float s    = cosf(mag);
    const float coef = sinf(mag) / mag;
#pragma unroll
    for (int i = 0; i < 6; ++i) t[i] *= coef;

    float* vv = v + (size_t)idx * 4;
    float o1, o2, o3, o4;
    apply_rotor(s, t, vv[0], vv[1], vv[2], vv[3], o1, o2, o3, o4);
    vv[0] = o1; vv[1] = o2; vv[2] = o3; vv[3] = o4;
}

// ---------------------------------------------------------------------------
// In-place cumsum over the sequence axis: one workgroup per (b,k) lane.
// Per-thread serial chunk of L/256 steps, float4 Hillis-Steele scan in LDS,
// then a second pass writes the inclusive prefix.
// ---------------------------------------------------------------------------
__global__ __launch_bounds__(256)
void cumsum_time(float* __restrict__ data, int Ln, int Kn, int Dn) {
    __shared__ float4 sm[256];
    const int blk = blockIdx.x;      // b*Kn + k
    const int b   = blk / Kn;
    const int k   = blk % Kn;
    const size_t base = (size_t)b * Ln * Dn + (size_t)k * 4;
    const int t  = threadIdx.x;
    const int CH = Ln / 256;         // 16

    float4 sum = {0.f, 0.f, 0.f, 0.f};
    for (int i = 0; i < CH; ++i) {
        const size_t off = base + (size_t)(t * CH + i) * Dn;
        sum.x += data[off + 0]; sum.y += data[off + 1];
        sum.z += data[off + 2]; sum.w += data[off + 3];
    }
    sm[t] = sum;
    __syncthreads();

    for (int off = 1; off < 256; off <<= 1) {
        float4 o = {0.f, 0.f, 0.f, 0.f};
        if (t >= off) o = sm[t - off];
        __syncthreads();
        sm[t].x += o.x; sm[t].y += o.y; sm[t].z += o.z; sm[t].w += o.w;
        __syncthreads();
    }

    float4 run = {0.f, 0.f, 0.f, 0.f};
    if (t > 0) run = sm[t - 1];      // exclusive prefix of this chunk
    for (int i = 0; i < CH; ++i) {
        const size_t off = base + (size_t)(t * CH + i) * Dn;
        run.x += data[off + 0]; run.y += data[off + 1];
        run.z += data[off + 2]; run.w += data[off + 3];
        data[off + 0] = run.x; data[off + 1] = run.y;
        data[off + 2] = run.z; data[off + 3] = run.w;
    }
}

// ---------------------------------------------------------------------------
// Reversed query rotor on memory, 1/sqrt(l+1) scaling, LayerNorm (affine).
// One workgroup per (b,l) row; thread k owns k-group of 4 channels.
// ---------------------------------------------------------------------------
__global__ __launch_bounds__(256)
void retrieve_ln(const float* __restrict__ angq, const float* __restrict__ mem,
                 const float* __restrict__ g, const float* __restrict__ bta,
                 float* __restrict__ rn, int Ln, int Kn, int Dn) {
    __shared__ float sS[256];
    __shared__ float sQ[256];
    const int row = blockIdx.x;      // b*Ln + l
    const int l   = row % Ln;
    const int k   = threadIdx.x;     // Kn == 256

    const float* a = angq + ((size_t)row * Kn + k) * 6;
    float t[6];
    float m2 = 0.f;
#pragma unroll
    for (int i = 0; i < 6; ++i) { const float ai = a[i]; t[i] = ai; m2 += ai * ai; }
    const float mag  = fmaxf(sqrtf(m2), 1e-8f);
    const float s    = cosf(mag);
    const float coef = -sinf(mag) / mag;  // reversed rotor: -biv
#pragma unroll
    for (int i = 0; i < 6; ++i) t[i] *= coef;

    const float* mv = mem + (size_t)row * Dn + (size_t)k * 4;
    float r1, r2, r3, r4;
    apply_rotor(s, t, mv[0], mv[1], mv[2], mv[3], r1, r2, r3, r4);
    const float invn = rsqrtf((float)(l + 1));
    r1 *= invn; r2 *= invn; r3 *= invn; r4 *= invn;

    sS[k] = r1 + r2 + r3 + r4;
    sQ[k] = r1 * r1 + r2 * r2 + r3 * r3 + r4 * r4;
    __syncthreads();
    for (int off = 128; off > 0; off >>= 1) {
        if (k < off) { sS[k] += sS[k + off]; sQ[k] += sQ[k + off]; }
        __syncthreads();
    }
    const float invD = 1.0f / (float)Dn;
    const float mean = sS[0] * invD;
    const float var  = sQ[0] * invD - mean * mean;
    const float rstd = rsqrtf(var + 1e-5f);

    const int d = k * 4;
    float* out = rn + (size_t)row * Dn + d;
    out[0] = (r1 - mean) * rstd * g[d + 0] + bta[d + 0];
    out[1] = (r2 - mean) * rstd * g[d + 1] + bta[d + 1];
    out[2] = (r3 - mean) * rstd * g[d + 2] + bta[d + 2];
    out[3] = (r4 - mean) * rstd * g[d + 3] + bta[d + 3];
}

// ---------------------------------------------------------------------------
// Launch pipeline
// ---------------------------------------------------------------------------
extern "C" void kernel_launch(void* const* d_in, const int* in_sizes, int n_in,
                              void* d_out, int out_size, void* d_ws, size_t ws_size,
                              hipStream_t stream) {
    (void)in_sizes; (void)n_in; (void)out_size; (void)ws_size;

    const float* x    = (const float*)d_in[0];
    const float* wk1  = (const float*)d_in[1];
    const float* bk1  = (const float*)d_in[2];
    const float* wk2  = (const float*)d_in[3];
    const float* bk2  = (const float*)d_in[4];
    const float* wq1  = (const float*)d_in[5];
    const float* bq1  = (const float*)d_in[6];
    const float* wq2  = (const float*)d_in[7];
    const float* bq2  = (const float*)d_in[8];
    const float* wv   = (const float*)d_in[9];
    const float* bv   = (const float*)d_in[10];
    const float* ln_g = (const float*)d_in[11];
    const float* ln_b = (const float*)d_in[12];
    const float* wo   = (const float*)d_in[13];
    const float* bo   = (const float*)d_in[14];
    float* out = (float*)d_out;

    const int Bn = 2, Ln = 4096, Dn = 1024, Kn = 256;
    const int Mrows = Bn * Ln;   // 8192
    const int N6    = Kn * 6;    // 1536

    // Workspace (fp32): h 32MB | angk 48MB | angq 48MB | rv 32MB = 160MB.
    float* h    = (float*)d_ws;                       // hidden, later reused for rn
    float* angk = h + (size_t)Mrows * Dn;
    float* angq = angk + (size_t)Mrows * N6;
    float* rv   = angq + (size_t)Mrows * N6;          // value -> rotated -> cumsum

    const dim3 blk(128);
    const dim3 gD(Dn / 64, Mrows / 64);   // 16 x 128
    const dim3 g6(N6 / 64, Mrows / 64);   // 24 x 128

    // key path
    gemm_wmma_f32<ACT_GELU,   false><<<gD, blk, 0, stream>>>(x, wk1, bk1, nullptr, h,    Mrows, Dn, Dn);
    gemm_wmma_f32<ACT_TANHPI, false><<<g6, blk, 0, stream>>>(h, wk2, bk2, nullptr, angk, Mrows, N6, Dn);
    // query path (reuses h)
    gemm_wmma_f32<ACT_GELU,   false><<<gD, blk, 0, stream>>>(x, wq1, bq1, nullptr, h,    Mrows, Dn, Dn);
    gemm_wmma_f32<ACT_TANHPI, false><<<g6, blk, 0, stream>>>(h, wq2, bq2, nullptr, angq, Mrows, N6, Dn);
    // value
    gemm_wmma_f32<ACT_NONE,   false><<<gD, blk, 0, stream>>>(x, wv, bv, nullptr, rv, Mrows, Dn, Dn);

    // key rotor on value (in place), then sequence cumsum (in place)
    const int total = Mrows * Kn;  // 2,097,152
    rotor_key_apply<<<dim3(total / 256), dim3(256), 0, stream>>>(angk, rv, total);
    cumsum_time<<<dim3(Bn * Kn), dim3(256), 0, stream>>>(rv, Ln, Kn, Dn);

    // reversed query rotor + 1/sqrt(pos) + LayerNorm -> rn (reuse h)
    retrieve_ln<<<dim3(Mrows), dim3(256), 0, stream>>>(angq, rv, ln_g, ln_b, h, Ln, Kn, Dn);

    // out = x + rn @ wo + bo
    gemm_wmma_f32<ACT_NONE, true><<<gD, blk, 0, stream>>>(h, wo, bo, x, out, Mrows, Dn, Dn);
}